// YoloLoss_59339268162225
// MI455X (gfx1250) — compile-verified
//
#include <hip/hip_runtime.h>

// ---------------- problem constants (from reference) ----------------
#define S_DIM        52
#define NBOX         3
#define NC           80
#define BOX_STRIDE   85            // 5 + NC
#define CELL_STRIDE  255           // NBOX * BOX_STRIDE
#define CELLS_PER_IMG (S_DIM * S_DIM)   // 2704
#define N_IMG        64
#define EPS_F        1e-6f
#define IOU_THRES_F  0.7f
// scale_idx = SCALES.index(52/13 = 4.0) = 2  -> anchors flat offset 2*3*2 = 12
#define ANC_OFFSET   12
#define OUT_ELEMS    257           // 4*64 + 1 (prior is always zero for epoch>=0)

typedef float v2f __attribute__((ext_vector_type(2)));
typedef float v8f __attribute__((ext_vector_type(8)));

#if __has_builtin(__builtin_amdgcn_wmma_f32_16x16x4_f32)
#define HAVE_WMMA_F32X4 1
#else
#define HAVE_WMMA_F32X4 0
#endif

__global__ void zero_out_kernel(float* __restrict__ out, int n) {
    int i = blockIdx.x * blockDim.x + threadIdx.x;
    if (i < n) out[i] = 0.0f;
}

__global__ __launch_bounds__(256)
void yolo_loss_kernel(const float* __restrict__ yhat,
                      const float* __restrict__ y,
                      const float* __restrict__ anchors,
                      float* __restrict__ out)
{
    const int n         = blockIdx.y;                       // image index
    const int cellInImg = blockIdx.x * 256 + threadIdx.x;   // 0..2703 (+pad)
    const bool active   = (cellInImg < CELLS_PER_IMG);

    float pc = 0.0f, pcl = 0.0f, pno = 0.0f, pob = 0.0f;

    if (active) {
        const int gy = cellInImg / S_DIM;
        const int gx = cellInImg - gy * S_DIM;
        const size_t base = ((size_t)n * CELLS_PER_IMG + cellInImg) * CELL_STRIDE;
        const float* __restrict__ hp = yhat + base;
        const float* __restrict__ tp = y + base;

        // ---- load box params (x,y,w,h,conf) for 3 pred + 3 true boxes ----
        float hx[NBOX], hy[NBOX], hw[NBOX], hh[NBOX], hcf[NBOX];
        float tx[NBOX], ty[NBOX], tw[NBOX], th[NBOX], tcf[NBOX];
        #pragma unroll
        for (int b = 0; b < NBOX; ++b) {
            const float* p = hp + b * BOX_STRIDE;
            hx[b] = p[0]; hy[b] = p[1]; hw[b] = p[2]; hh[b] = p[3]; hcf[b] = p[4];
            const float* q = tp + b * BOX_STRIDE;
            tx[b] = q[0]; ty[b] = q[1]; tw[b] = q[2]; th[b] = q[3]; tcf[b] = q[4];
        }

        // ---- yolo2bbox corners (absolute, normalized) ----
        const float invS = 1.0f / (float)S_DIM;
        const float fgx = (float)gx, fgy = (float)gy;
        float ax1[NBOX], ay1[NBOX], ax2[NBOX], ay2[NBOX], aA[NBOX];
        float bx1[NBOX], by1[NBOX], bx2[NBOX], by2[NBOX], bA[NBOX];
        #pragma unroll
        for (int b = 0; b < NBOX; ++b) {
            float cx = (hx[b] + fgx) * invS, cy = (hy[b] + fgy) * invS;
            ax1[b] = cx - hw[b] * 0.5f; ax2[b] = cx + hw[b] * 0.5f;
            ay1[b] = cy - hh[b] * 0.5f; ay2[b] = cy + hh[b] * 0.5f;
            aA[b]  = (ax2[b] - ax1[b]) * (ay2[b] - ay1[b]);
            cx = (tx[b] + fgx) * invS;  cy = (ty[b] + fgy) * invS;
            bx1[b] = cx - tw[b] * 0.5f; bx2[b] = cx + tw[b] * 0.5f;
            by1[b] = cy - th[b] * 0.5f; by2[b] = cy + th[b] * 0.5f;
            bA[b]  = (bx2[b] - bx1[b]) * (by2[b] - by1[b]);
        }

        // ---- 3x3 IoU ----
        float iou[NBOX][NBOX];
        #pragma unroll
        for (int i = 0; i < NBOX; ++i) {
            #pragma unroll
            for (int j = 0; j < NBOX; ++j) {
                float wi = fmaxf(fminf(ax2[i], bx2[j]) - fmaxf(ax1[i], bx1[j]), 0.0f);
                float hi = fmaxf(fminf(ay2[i], by2[j]) - fmaxf(ay1[i], by1[j]), 0.0f);
                float inter = wi * hi;
                float uni = aA[i] + bA[j] - inter;
                iou[i][j] = inter / (uni + EPS_F);
            }
        }

        // max over true boxes (per predicted box) -> no-obj mask
        float noobj_m[NBOX];
        #pragma unroll
        for (int i = 0; i < NBOX; ++i) {
            float m = fmaxf(iou[i][0], fmaxf(iou[i][1], iou[i][2]));
            noobj_m[i] = (m < IOU_THRES_F) ? 1.0f : 0.0f;
        }

        // argmax over predicted boxes (per true box), first-max semantics
        int idx[NBOX];
        #pragma unroll
        for (int t = 0; t < NBOX; ++t) {
            int bi = 0; float bv = iou[0][t];
            if (iou[1][t] > bv) { bv = iou[1][t]; bi = 1; }
            if (iou[2][t] > bv) { bv = iou[2][t]; bi = 2; }
            idx[t] = bi;
        }

        // ---- wh log-space vs anchors (scale_idx = 2) ----
        float whw_h[NBOX], whh_h[NBOX], whw_t[NBOX], whh_t[NBOX];
        #pragma unroll
        for (int b = 0; b < NBOX; ++b) {
            float aw = anchors[ANC_OFFSET + 2 * b + 0];
            float ah = anchors[ANC_OFFSET + 2 * b + 1];
            whw_h[b] = logf(hw[b] / aw + EPS_F);
            whh_h[b] = logf(hh[b] / ah + EPS_F);
            whw_t[b] = logf(tw[b] / aw + EPS_F);
            whh_t[b] = logf(th[b] / ah + EPS_F);
        }

        float have[NBOX];
        #pragma unroll
        for (int t = 0; t < NBOX; ++t) have[t] = (tcf[t] > 0.0f) ? 1.0f : 0.0f;

        // ---- coord + obj losses ----
        #pragma unroll
        for (int t = 0; t < NBOX; ++t) {
            int it = idx[t];
            float sx = (it == 0) ? hx[0] : ((it == 1) ? hx[1] : hx[2]);
            float sy = (it == 0) ? hy[0] : ((it == 1) ? hy[1] : hy[2]);
            float sw = (it == 0) ? whw_h[0] : ((it == 1) ? whw_h[1] : whw_h[2]);
            float sh = (it == 0) ? whh_h[0] : ((it == 1) ? whh_h[1] : whh_h[2]);
            float sc = (it == 0) ? hcf[0] : ((it == 1) ? hcf[1] : hcf[2]);
            float d0 = sx - tx[t], d1 = sy - ty[t];
            float d2 = sw - whw_t[t], d3 = sh - whh_t[t];
            float s = (d0 * d0 + d1 * d1 + d2 * d2 + d3 * d3) * have[t];
            s *= (2.0f - tw[t] * th[t]);            // SCALE_COORD
            pc += s;
            float dc = sc - tcf[t];
            pob += dc * dc * have[t];
        }

        // ---- no-obj loss (target 0, NO_OBJ_V3) ----
        #pragma unroll
        for (int b = 0; b < NBOX; ++b) pno += hcf[b] * hcf[b] * noobj_m[b];

        // ---- class loss: selection hoisted into the address (no per-class
        //      cndmask chains); per-t loads stay contiguous -> b128 merge ----
        const int off0 = idx[0] * BOX_STRIDE + 5;
        const int off1 = idx[1] * BOX_STRIDE + 5;
        const int off2 = idx[2] * BOX_STRIDE + 5;
        float cl0 = 0.0f, cl1 = 0.0f, cl2 = 0.0f;
        #pragma unroll 4
        for (int c = 0; c < NC; ++c) {
            float s0 = hp[off0 + c] - tp[0 * BOX_STRIDE + 5 + c];
            float s1 = hp[off1 + c] - tp[1 * BOX_STRIDE + 5 + c];
            float s2 = hp[off2 + c] - tp[2 * BOX_STRIDE + 5 + c];
            cl0 += s0 * s0; cl1 += s1 * s1; cl2 += s2 * s2;
        }
        pcl = cl0 * have[0] + cl1 * have[1] + cl2 * have[2];
    }

    // ================= block reduction =================
    // 1) intra-wave butterfly (wave32)
    float v0 = pc, v1 = pcl, v2 = pno, v3 = pob;
    #pragma unroll
    for (int m = 16; m >= 1; m >>= 1) {
        v0 += __shfl_xor(v0, m, 32);
        v1 += __shfl_xor(v1, m, 32);
        v2 += __shfl_xor(v2, m, 32);
        v3 += __shfl_xor(v3, m, 32);
    }
    __shared__ float red[32];   // 8 waves x 4 components
    const int lane = threadIdx.x & 31;
    const int wv   = threadIdx.x >> 5;
    if (lane == 0) {
        red[wv * 4 + 0] = v0; red[wv * 4 + 1] = v1;
        red[wv * 4 + 2] = v2; red[wv * 4 + 3] = v3;
    }
    __syncthreads();

    // 2) wave 0: cross-wave sum via fp32 WMMA (exact, RNE):
    //    D = A(16x4) * ones(4x16) + 0  ->  D[m][*] = row-sum of 4 wave-partials.
    //    Component c uses rows 2c (waves 0..3) and 2c+1 (waves 4..7).
    if (threadIdx.x < 32) {
#if HAVE_WMMA_F32X4
        const int l  = threadIdx.x;
        const int mm = l & 15;
        v2f a; a.x = 0.0f; a.y = 0.0f;
        if (mm < 8) {
            const int c  = mm >> 1;
            const int r  = mm & 1;
            const int w0 = 4 * r + ((l < 16) ? 0 : 2);   // A-layout: lanes 0-15 hold K=0,1; 16-31 hold K=2,3
            a.x = red[w0 * 4 + c];
            a.y = red[(w0 + 1) * 4 + c];
        }
        v2f bm; bm.x = 1.0f; bm.y = 1.0f;                // all-ones B: layout-independent
        v8f cacc = {};
        v8f d = __builtin_amdgcn_wmma_f32_16x16x4_f32(
            /*neg_a=*/false, a, /*neg_b=*/false, bm,
            /*c_mod=*/(short)0, cacc, /*reuse_a=*/false, /*reuse_b=*/false);
        if (l == 0) {   // lane 0 holds D[m][0] in d[m], m = 0..7
            atomicAdd(&out[0 * N_IMG + n], d[0] + d[1]);  // coord
            atomicAdd(&out[1 * N_IMG + n], d[2] + d[3]);  // class
            atomicAdd(&out[2 * N_IMG + n], d[4] + d[5]);  // no-obj
            atomicAdd(&out[3 * N_IMG + n], d[6] + d[7]);  // obj
        }
#else
        // fallback: lane c sums its component across the 8 waves
        if (threadIdx.x < 4) {
            float s = 0.0f;
            #pragma unroll
            for (int w = 0; w < 8; ++w) s += red[w * 4 + threadIdx.x];
            atomicAdd(&out[threadIdx.x * N_IMG + n], s);
        }
#endif
    }
    // out[256] (prior loss) stays 0 from zero_out_kernel: epoch>=EPOCH_PRIOR(=0)
    // in the harness config, and LAMBDA_PRIOR==0 regardless.
}

extern "C" void kernel_launch(void* const* d_in, const int* in_sizes, int n_in,
                              void* d_out, int out_size, void* d_ws, size_t ws_size,
                              hipStream_t stream) {
    const float* yhat    = (const float*)d_in[0];
    const float* y       = (const float*)d_in[1];
    const float* anchors = (const float*)d_in[2];
    // d_in[3] = epoch (int, ==1): prior term is identically zero -> unused.
    float* out = (float*)d_out;

    zero_out_kernel<<<(OUT_ELEMS + 255) / 256, 256, 0, stream>>>(out, OUT_ELEMS);

    dim3 grid((CELLS_PER_IMG + 255) / 256, N_IMG);   // 11 x 64 blocks
    yolo_loss_kernel<<<grid, 256, 0, stream>>>(yhat, y, anchors, out);
}